// SimonCipher_56942676410720
// MI455X (gfx1250) — compile-verified
//
#include <hip/hip_runtime.h>

// Soft-Simon cipher: elementwise, 32 rounds, transcendental-bound.
// CDNA5 paths used: global_load_async_to_lds_b64/b128 (ASYNCcnt) for input
// staging, s_wait_asynccnt, fast v_exp_f32 / v_rcp_f32 lowering of sigmoid.

#define TPB 256
#define EPT 2          // independent elements per thread (ILP for TRANS latency)
#define NROUNDS 32

// sigmoid(10*(v-0.5)) = 1 / (1 + exp2((0.5 - v) * 10 * log2(e)))
__device__ __forceinline__ float sharpf(float v) {
  float e = __builtin_amdgcn_exp2f(fmaf(v, -14.426950408889634f, 7.213475204444817f));
  return __builtin_amdgcn_rcpf(1.0f + e);
}

// soft_xor on already-sharpened inputs
__device__ __forceinline__ float soft_xor_pre(float as, float bs) {
  float t1 = as * (1.0f - bs);
  float t2 = (1.0f - as) * bs;
  float r  = fmaf(-t1, t2, t1 + t2);
  return fminf(fmaxf(r, 0.0f), 1.0f);
}

__device__ __forceinline__ float mod65536f(float a) {
  return fmaf(-floorf(a * (1.0f / 65536.0f)), 65536.0f, a);
}

// rotate_left(x, n): mul_l = 2^n, mul_r = 2^-(16-n)
__device__ __forceinline__ float rotlf(float x, float mul_l, float mul_r) {
  float xs = x * 65536.0f;
  float sl = mod65536f(xs * mul_l);
  float sr = xs * mul_r;
  return mod65536f(sl + sr) * (1.0f / 65536.0f);
}

__global__ __launch_bounds__(TPB, 1) void simon_soft_kernel(
    const float2* __restrict__ pt, const float4* __restrict__ key,
    float2* __restrict__ out, int n) {
  __shared__ float2 s_pt[TPB * EPT];
  __shared__ float4 s_key[TPB * EPT];

  const int tid  = threadIdx.x;
  const int base = blockIdx.x * (TPB * EPT);

  // ---- CDNA5 async global->LDS staging (tracked with ASYNCcnt) ----
#pragma unroll
  for (int e = 0; e < EPT; ++e) {
    const int li = tid + e * TPB;
    const int gi = base + li;
    if (gi < n) {
      unsigned lds_pt = (unsigned)(unsigned long long)(&s_pt[li]);
      unsigned lds_k  = (unsigned)(unsigned long long)(&s_key[li]);
      unsigned long long ga_pt = (unsigned long long)(pt + gi);
      unsigned long long ga_k  = (unsigned long long)(key + gi);
      asm volatile("global_load_async_to_lds_b64 %0, %1, off"
                   :: "v"(lds_pt), "v"(ga_pt) : "memory");
      asm volatile("global_load_async_to_lds_b128 %0, %1, off"
                   :: "v"(lds_k), "v"(ga_k) : "memory");
    }
  }
  asm volatile("s_wait_asynccnt 0" ::: "memory");
  __syncthreads();

  float x[EPT], y[EPT], ks0[EPT], ks1[EPT], ks2[EPT], ks3[EPT];
  int gi_[EPT];
#pragma unroll
  for (int e = 0; e < EPT; ++e) {
    const int li = tid + e * TPB;
    gi_[e] = base + li;
    float2 p  = s_pt[li];
    float4 kk = s_key[li];
    x[e] = p.x;
    y[e] = p.y;
    // round keys repeat with period 4: pre-sharpen once (saves 28 sigmoids)
    ks0[e] = sharpf(kk.x);
    ks1[e] = sharpf(kk.y);
    ks2[e] = sharpf(kk.z);
    ks3[e] = sharpf(kk.w);
  }

  // outer loop kept rolled (8 iters) to bound I-cache; inner 4 rounds unrolled
  // so the (round % 4) key index becomes a register, EPT unrolled for ILP.
  for (int rr = 0; rr < NROUNDS / 4; ++rr) {
#pragma unroll
    for (int j = 0; j < 4; ++j) {
#pragma unroll
      for (int e = 0; e < EPT; ++e) {
        const float ks = (j == 0) ? ks0[e]
                       : (j == 1) ? ks1[e]
                       : (j == 2) ? ks2[e]
                                  : ks3[e];
        float s1 = rotlf(x[e], 2.0f,   1.0f / 32768.0f);
        float s8 = rotlf(x[e], 256.0f, 1.0f / 256.0f);
        float s2 = rotlf(x[e], 4.0f,   1.0f / 16384.0f);
        float at = sharpf(s1) * sharpf(s8);
        float fx = soft_xor_pre(sharpf(at), sharpf(s2));
        float u  = soft_xor_pre(sharpf(y[e]), sharpf(fx));
        float xn = soft_xor_pre(sharpf(u), ks);
        y[e] = x[e];
        x[e] = xn;
      }
    }
  }

#pragma unroll
  for (int e = 0; e < EPT; ++e) {
    if (gi_[e] < n) {
      float2 o;
      o.x = x[e];
      o.y = y[e];
      out[gi_[e]] = o;
    }
  }
}

extern "C" void kernel_launch(void* const* d_in, const int* in_sizes, int n_in,
                              void* d_out, int out_size, void* d_ws, size_t ws_size,
                              hipStream_t stream) {
  (void)n_in; (void)d_ws; (void)ws_size; (void)out_size;
  const float2* pt  = (const float2*)d_in[0];   // (BATCH, 2) float32
  const float4* key = (const float4*)d_in[1];   // (BATCH, 4) float32
  float2* out = (float2*)d_out;                 // (BATCH, 2) float32
  const int n = in_sizes[0] / 2;                // BATCH
  const int per_block = TPB * EPT;
  const int blocks = (n + per_block - 1) / per_block;
  simon_soft_kernel<<<blocks, TPB, 0, stream>>>(pt, key, out, n);
}